// VNLeakyReLU_68083821576959
// MI455X (gfx1250) — compile-verified
//
#include <hip/hip_runtime.h>

typedef __attribute__((ext_vector_type(2))) float v2f;
typedef __attribute__((ext_vector_type(8))) float v8f;

namespace {
constexpr int   C         = 64;      // channels (in == out)
constexpr int   NN        = 32768;   // spatial
constexpr int   WGN       = 64;      // n-columns per workgroup
constexpr float NEG_SLOPE = 0.2f;
constexpr float EPS       = 1e-6f;
}

// Fused VNLinear (64x64 GEMM via V_WMMA_F32_16X16X4_F32) + VN leaky ReLU.
// Block = 128 threads = 4 waves; each wave owns a 16-row o-tile.
// LDS fill uses CDNA5 async global->LDS DMA (ASYNCcnt), no VGPR staging.
__global__ __launch_bounds__(128) void vn_leaky_wmma(
    const float* __restrict__ x,   // [B][C][3][N]
    const float* __restrict__ W,   // [C][C]
    float* __restrict__ out)       // [B][C][3][N]
{
    __shared__ float Xs[3 * C * WGN];   // [v][i][n]  48 KB

    const int tid  = threadIdx.x;
    const int blk  = blockIdx.x;
    const int nblk = NN / WGN;           // 512
    const int b    = blk / nblk;
    const int n0   = (blk % nblk) * WGN;

    const size_t bstride = (size_t)C * 3 * NN;
    const float* xb  = x   + (size_t)b * bstride + n0;
    float*       ob  = out + (size_t)b * bstride + n0;

    // ---- async LDS fill: 3*64*64 floats, 16B per lane per issue.
    // LDS dest address = low 32 bits of the flat address of &Xs[...]
    // (flat LDS aperture: addr[31:0] is the wave-relative LDS byte address).
    // Passing ptrtoint(&Xs[...]) into the asm also makes Xs escape, so the
    // compiler must assume the asm writes it (fixes undef-fold of Xs reads).
    constexpr int E4 = 3 * C * WGN / 4;         // 3072 float4
    for (int e4 = tid; e4 < E4; e4 += 128) {
        const int v  = e4 / (C * WGN / 4);
        const int r  = e4 % (C * WGN / 4);
        const int i  = r / (WGN / 4);
        const int n4 = r % (WGN / 4);
        const float*   gp  = xb + (size_t)i * 3 * NN + (size_t)v * NN + n4 * 4;
        const unsigned lds = (unsigned)(unsigned long long)&Xs[e4 * 4];
        asm volatile("global_load_async_to_lds_b128 %0, %1, off"
                     :: "v"(lds), "v"(gp) : "memory");
    }
    asm volatile("s_wait_asynccnt 0x0" ::: "memory");
    __syncthreads();

    const int wave = tid >> 5;
    const int lane = tid & 31;
    const int o0   = wave * 16;        // this wave's output-channel tile
    const int m    = lane & 15;        // M row (A) / N col (B,C,D)
    const int kh   = (lane >> 4) * 2;  // K phase for A/B operand halves: 0 or 2
    const int oh   = (lane >> 4) * 8;  // M offset for C/D halves

    // ---- A operand: W[o0+m][*] in 16 chunks of K=4 (v2f per chunk) ----
    v2f A[16];
    const float* wrow = W + (o0 + m) * C;
#pragma unroll
    for (int q = 0; q < 16; ++q)
        A[q] = *(const v2f*)(wrow + 4 * q + kh);   // 8B-aligned (4q+kh even)

    for (int sub = 0; sub < WGN / 16; ++sub) {
        const int nl = sub * 16 + m;   // n within LDS tile for this lane

        // ---- GEMM: acc[v] = W[o-tile,:] * X[:, v, n-tile] ----
        v8f acc[3] = {};
#pragma unroll
        for (int q = 0; q < 16; ++q) {
            const int krow = 4 * q + kh;
#pragma unroll
            for (int v = 0; v < 3; ++v) {
                v2f Bop;
                Bop.x = Xs[(v * C + krow    ) * WGN + nl];
                Bop.y = Xs[(v * C + krow + 1) * WGN + nl];
                acc[v] = __builtin_amdgcn_wmma_f32_16x16x4_f32(
                    /*neg_a=*/false, A[q], /*neg_b=*/false, Bop,
                    /*c_mod=*/(short)0, acc[v],
                    /*reuse_a=*/false, /*reuse_b=*/false);
            }
        }

        // ---- VN leaky ReLU on the 16x16 D tile (per C/D VGPR layout) ----
#pragma unroll
        for (int j = 0; j < 8; ++j) {
            const int o = o0 + j + oh;           // absolute output channel
            float xv[3], dv[3];
#pragma unroll
            for (int v = 0; v < 3; ++v) {
                xv[v] = Xs[(v * C + o) * WGN + nl];
                dv[v] = acc[v][j];
            }
            const float dot = xv[0]*dv[0] + xv[1]*dv[1] + xv[2]*dv[2];
            const float dns = dv[0]*dv[0] + dv[1]*dv[1] + dv[2]*dv[2];
            const float s   = (dot >= 0.0f)
                                ? 0.0f
                                : (1.0f - NEG_SLOPE) * dot / (dns + EPS);
#pragma unroll
            for (int v = 0; v < 3; ++v)
                ob[(size_t)o * 3 * NN + (size_t)v * NN + nl] = xv[v] - s * dv[v];
        }
        // Xs is read-only after the fill; no barrier needed between subtiles.
    }
}

extern "C" void kernel_launch(void* const* d_in, const int* in_sizes, int n_in,
                              void* d_out, int out_size, void* d_ws, size_t ws_size,
                              hipStream_t stream)
{
    const float* x = (const float*)d_in[0];
    const float* W = (const float*)d_in[1];
    float*     out = (float*)d_out;

    const int B = in_sizes[0] / (C * 3 * NN);     // 4 for the reference shapes
    dim3 grid(B * (NN / WGN));                    // 2048 workgroups
    vn_leaky_wmma<<<grid, 128, 0, stream>>>(x, W, out);
}